// GIN_25237227832045
// MI455X (gfx1250) — compile-verified
//
#include <hip/hip_runtime.h>

// ---------------------------------------------------------------------------
// GIN inference for MI455X (gfx1250, wave32, WMMA).
// Dense matmuls use V_WMMA_F32_16X16X4_F32 (fp32 end-to-end, matching ref).
// Aggregation is pushed AFTER the first Linear of each MLP (linear commutes
// with segment_sum), so all scatters are 32-channel: 4x less atomic traffic
// on layer 1.
// ---------------------------------------------------------------------------

typedef float v2f __attribute__((ext_vector_type(2)));
typedef float v8f __attribute__((ext_vector_type(8)));

#define NNODES 100000
#define NEDGES 2500000
#define DIN    128
#define HID    32
#define NGRAPH 512
#define NOUT   2
#define MTILES (NNODES / 16)      // 6250 exact
#define BLOCK  256                 // 8 waves of 32

// D = A(16x4) * B(4x16) + C   (fp32 WMMA)
static __device__ __forceinline__ v8f wmma_f32(v2f a, v2f b, v8f c) {
  return __builtin_amdgcn_wmma_f32_16x16x4_f32(false, a, false, b, (short)0, c,
                                               false, false);
}

// Y = X[N,K] @ W[K,32]; also duplicate into Z (self term for GIN aggregation).
// One wave per 16-row tile; two 16x16 accumulators cover all 32 out channels.
template <int K>
__global__ __launch_bounds__(BLOCK) void k_linear(const float* __restrict__ X,
                                                  const float* __restrict__ W,
                                                  float* __restrict__ Y,
                                                  float* __restrict__ Z) {
  __shared__ float wlds[K * 32];
  for (int i = threadIdx.x; i < K * 32; i += BLOCK) wlds[i] = W[i];
  __syncthreads();

  const int tile = (int)((blockIdx.x * BLOCK + threadIdx.x) >> 5);
  const int lane = threadIdx.x & 31;
  if (tile >= MTILES) return;           // uniform per wave: EXEC stays full
  const int ml = lane & 15;             // A: row, B: col, C: col
  const int kh = lane >> 4;             // which K-pair this lane holds

  const float* xrow = X + (size_t)(tile * 16 + ml) * K + 2 * kh;
  v8f c0 = {0.f, 0.f, 0.f, 0.f, 0.f, 0.f, 0.f, 0.f};
  v8f c1 = {0.f, 0.f, 0.f, 0.f, 0.f, 0.f, 0.f, 0.f};

#pragma unroll 8
  for (int kb = 0; kb < K; kb += 4) {
    v2f a = *(const v2f*)(xrow + kb);               // (M=ml, K=kb+2kh, +1)
    const float* w = wlds + (kb + 2 * kh) * 32;     // B rows kb+2kh, +1
    v2f b0 = {w[ml],      w[32 + ml]};              // cols 0..15
    v2f b1 = {w[16 + ml], w[48 + ml]};              // cols 16..31
    c0 = wmma_f32(a, b0, c0);
    c1 = wmma_f32(a, b1, c1);
  }

  const int row0 = tile * 16 + kh * 8;              // C layout: M = r + 8*kh
#pragma unroll
  for (int r = 0; r < 8; ++r) {
    size_t o = (size_t)(row0 + r) * 32 + ml;
    float v0 = c0[r], v1 = c1[r];
    Y[o] = v0; Y[o + 16] = v1;
    Z[o] = v0; Z[o + 16] = v1;
  }
}

// H = relu( relu(BN(Zb + b1)) @ W2 + b2 ).  BN folded to t = relu(z*s + d).
__global__ __launch_bounds__(BLOCK) void k_mlp2(
    const float* __restrict__ Zb, const float* __restrict__ b1,
    const float* __restrict__ gamma, const float* __restrict__ beta,
    const float* __restrict__ rm, const float* __restrict__ rv,
    const float* __restrict__ W2, const float* __restrict__ b2,
    float* __restrict__ H) {
  __shared__ float wlds[32 * 32];
  __shared__ float sc[32], sh[32];
  for (int i = threadIdx.x; i < 1024; i += BLOCK) wlds[i] = W2[i];
  if (threadIdx.x < 32) {
    int c = threadIdx.x;
    float s = gamma[c] * rsqrtf(rv[c] + 1e-5f);
    sc[c] = s;
    sh[c] = (b1[c] - rm[c]) * s + beta[c];
  }
  __syncthreads();

  const int tile = (int)((blockIdx.x * BLOCK + threadIdx.x) >> 5);
  const int lane = threadIdx.x & 31;
  if (tile >= MTILES) return;
  const int ml = lane & 15;
  const int kh = lane >> 4;

  const float* zrow = Zb + (size_t)(tile * 16 + ml) * 32 + 2 * kh;
  v8f c0 = {0.f, 0.f, 0.f, 0.f, 0.f, 0.f, 0.f, 0.f};
  v8f c1 = {0.f, 0.f, 0.f, 0.f, 0.f, 0.f, 0.f, 0.f};

#pragma unroll
  for (int i = 0; i < 8; ++i) {
    const int k0 = 4 * i + 2 * kh;
    v2f z = *(const v2f*)(zrow + 4 * i);
    v2f a = {fmaxf(z.x * sc[k0]     + sh[k0],     0.f),
             fmaxf(z.y * sc[k0 + 1] + sh[k0 + 1], 0.f)};
    const float* w = wlds + k0 * 32;
    v2f b0 = {w[ml],      w[32 + ml]};
    v2f b1v = {w[16 + ml], w[48 + ml]};
    c0 = wmma_f32(a, b0, c0);
    c1 = wmma_f32(a, b1v, c1);
  }

  const float bias0 = b2[ml], bias1 = b2[16 + ml];
  const int row0 = tile * 16 + kh * 8;
#pragma unroll
  for (int r = 0; r < 8; ++r) {
    size_t o = (size_t)(row0 + r) * 32 + ml;
    H[o]      = fmaxf(c0[r] + bias0, 0.f);
    H[o + 16] = fmaxf(c1[r] + bias1, 0.f);
  }
}

// Z[dst] += Y[src] over 32 channels. One wave per edge: lanes share the edge
// index (broadcast SGPR-friendly load), gather/atomic are 128B coalesced.
__global__ __launch_bounds__(BLOCK) void k_scatter(const float* __restrict__ Y,
                                                   const int* __restrict__ ei,
                                                   float* __restrict__ Z) {
  const int idx = (int)(blockIdx.x * BLOCK + threadIdx.x);
  const int e = idx >> 5;
  if (e >= NEDGES) return;
  const int f = idx & 31;
  const int s = ei[e];
  const int d = ei[NEDGES + e];
  atomicAdd(&Z[(size_t)d * 32 + f], Y[(size_t)s * 32 + f]);
}

__global__ __launch_bounds__(BLOCK) void k_zero(float* __restrict__ p, int n) {
  int i = (int)(blockIdx.x * BLOCK + threadIdx.x);
  if (i < n) p[i] = 0.f;
}

// Gb[batch[n]] += H[n]  (global_add_pool)
__global__ __launch_bounds__(BLOCK) void k_pool(const float* __restrict__ H,
                                                const int* __restrict__ batch,
                                                float* __restrict__ Gb) {
  const int idx = (int)(blockIdx.x * BLOCK + threadIdx.x);
  const int n = idx >> 5;
  if (n >= NNODES) return;
  const int f = idx & 31;
  atomicAdd(&Gb[(size_t)batch[n] * 32 + f], H[(size_t)n * 32 + f]);
}

// out = relu(Gb @ lin1_W + lin1_b) @ lin2_W + lin2_b.  One wave per graph.
__global__ __launch_bounds__(32) void k_head(const float* __restrict__ Gb,
                                             const float* __restrict__ W1,
                                             const float* __restrict__ B1,
                                             const float* __restrict__ W2,
                                             const float* __restrict__ B2,
                                             float* __restrict__ out) {
  __shared__ float g[32], t[32];
  const int gid = blockIdx.x, j = threadIdx.x;
  g[j] = Gb[(size_t)gid * 32 + j];
  __syncthreads();
  float acc = B1[j];
#pragma unroll
  for (int k = 0; k < 32; ++k) acc = fmaf(g[k], W1[k * 32 + j], acc);
  t[j] = fmaxf(acc, 0.f);
  __syncthreads();
  if (j < NOUT) {
    float o = B2[j];
#pragma unroll
    for (int k = 0; k < 32; ++k) o = fmaf(t[k], W2[k * NOUT + j], o);
    out[(size_t)gid * NOUT + j] = o;
  }
}

extern "C" void kernel_launch(void* const* d_in, const int* in_sizes, int n_in,
                              void* d_out, int out_size, void* d_ws,
                              size_t ws_size, hipStream_t stream) {
  const float* x       = (const float*)d_in[0];
  const int*   ei      = (const int*)d_in[1];   // [2,E] flat: src | dst
  const int*   batch   = (const int*)d_in[2];
  const float* W1_0    = (const float*)d_in[3];
  const float* b1_0    = (const float*)d_in[4];
  const float* gamma_0 = (const float*)d_in[5];
  const float* beta_0  = (const float*)d_in[6];
  const float* rm_0    = (const float*)d_in[7];
  const float* rv_0    = (const float*)d_in[8];
  const float* W2_0    = (const float*)d_in[9];
  const float* b2_0    = (const float*)d_in[10];
  const float* W1s     = (const float*)d_in[11];
  const float* b1s     = (const float*)d_in[12];
  const float* gammas  = (const float*)d_in[13];
  const float* betas   = (const float*)d_in[14];
  const float* rms     = (const float*)d_in[15];
  const float* rvs     = (const float*)d_in[16];
  const float* W2s     = (const float*)d_in[17];
  const float* b2s     = (const float*)d_in[18];
  const float* lin1_W  = (const float*)d_in[19];
  const float* lin1_b  = (const float*)d_in[20];
  const float* lin2_W  = (const float*)d_in[21];
  const float* lin2_b  = (const float*)d_in[22];
  float* out = (float*)d_out;

  float* yb = (float*)d_ws;                 // [N,32] post-Linear1 features
  float* zb = yb + (size_t)NNODES * 32;     // [N,32] self + neighbor sum
  float* hb = zb + (size_t)NNODES * 32;     // [N,32] layer output
  float* gb = hb + (size_t)NNODES * 32;     // [G,32] pooled

  const int linblocks  = (MTILES + 7) / 8;                // 8 waves/block
  const int scatblocks = (NEDGES * 32 + BLOCK - 1) / BLOCK;
  const int poolblocks = (NNODES * 32 + BLOCK - 1) / BLOCK;

  // ---- layer 1 (D_IN=128 -> 32, aggregation moved after W1) ----
  k_linear<DIN><<<linblocks, BLOCK, 0, stream>>>(x, W1_0, yb, zb);
  k_scatter<<<scatblocks, BLOCK, 0, stream>>>(yb, ei, zb);
  k_mlp2<<<linblocks, BLOCK, 0, stream>>>(zb, b1_0, gamma_0, beta_0, rm_0,
                                          rv_0, W2_0, b2_0, hb);
  // ---- layers 2..5 (32 -> 32) ----
  for (int i = 0; i < 4; ++i) {
    k_linear<HID><<<linblocks, BLOCK, 0, stream>>>(hb, W1s + i * 1024, yb, zb);
    k_scatter<<<scatblocks, BLOCK, 0, stream>>>(yb, ei, zb);
    k_mlp2<<<linblocks, BLOCK, 0, stream>>>(zb, b1s + i * 32, gammas + i * 32,
                                            betas + i * 32, rms + i * 32,
                                            rvs + i * 32, W2s + i * 1024,
                                            b2s + i * 32, hb);
  }
  // ---- pool + head ----
  k_zero<<<(NGRAPH * 32 + BLOCK - 1) / BLOCK, BLOCK, 0, stream>>>(gb,
                                                                  NGRAPH * 32);
  k_pool<<<poolblocks, BLOCK, 0, stream>>>(hb, batch, gb);
  k_head<<<NGRAPH, 32, 0, stream>>>(gb, lin1_W, lin1_b, lin2_W, lin2_b, out);
}